// LightGCN_66357244723249
// MI455X (gfx1250) — compile-verified
//
#include <hip/hip_runtime.h>
#include <stdint.h>

#define N_USERS 100000
#define N_ITEMS 50000
#define N_NODES 150000
#define NNZ     10000000
#define D       64
#define N_HOPS  3
#define OUT_STRIDE (4 * D)           /* 256 floats per node in output */
#define EDGE_SCALE 2.0f              /* 1/(1-0.5)  */
#define MESS_SCALE (1.0f / 0.9f)     /* 1/(1-0.1)  */
#define TILE    2048                 /* edges staged per block */
#define THREADS 256
#define FULL_TILES (NNZ / TILE)      /* 4882 full tiles */
#define TAIL_START ((long)FULL_TILES * TILE) /* 9,998,336 */

#if defined(__AMDGCN__) && __has_builtin(__builtin_amdgcn_global_load_async_to_lds_b128)
#define USE_ASYNC_LDS 1
typedef int v4i_vec __attribute__((__vector_size__(4 * sizeof(int))));
typedef __attribute__((address_space(1))) v4i_vec* gptr_b128;
typedef __attribute__((address_space(3))) v4i_vec* lptr_b128;
#else
#define USE_ASYNC_LDS 0
#endif

// ---------------------------------------------------------------------------
// SpMM hop (full tiles only): out[:, hop+1, :] += edge_scaled(A) @ out[:, hop, :]
// Straight-line prologue: async-to-LDS DMA of a 2048-edge tile, wait ASYNCcnt,
// then each wave32 ballot-compacts surviving edges and does a cooperative
// gather(float2/lane) -> global_atomic_add_f32 scatter.
// ---------------------------------------------------------------------------
__global__ __launch_bounds__(THREADS) void lgcn_spmm(
    const int* __restrict__ rows, const int* __restrict__ cols,
    const float* __restrict__ vals, const unsigned char* __restrict__ emask,
    float* __restrict__ out, int hop) {
    __shared__ int   s_rows[TILE];
    __shared__ int   s_cols[TILE];
    __shared__ float s_vals[TILE];

    const long tileBase = (long)blockIdx.x * TILE;
    const int  tid      = threadIdx.x;

#if USE_ASYNC_LDS
    {
        // 8 KB per array = 512 x 16B chunks; 256 lanes x 2 chunks each.
        char* gR = (char*)(rows + tileBase);
        char* gC = (char*)(cols + tileBase);
        char* gV = (char*)(vals + tileBase);
#pragma unroll
        for (int k = 0; k < 2; ++k) {
            int off = (tid + k * THREADS) * 16;
            __builtin_amdgcn_global_load_async_to_lds_b128(
                (gptr_b128)(gR + off),
                (lptr_b128)((char*)s_rows + off), 0, 0);
            __builtin_amdgcn_global_load_async_to_lds_b128(
                (gptr_b128)(gC + off),
                (lptr_b128)((char*)s_cols + off), 0, 0);
            __builtin_amdgcn_global_load_async_to_lds_b128(
                (gptr_b128)(gV + off),
                (lptr_b128)((char*)s_vals + off), 0, 0);
        }
#if __has_builtin(__builtin_amdgcn_s_wait_asynccnt)
        __builtin_amdgcn_s_wait_asynccnt(0);
#else
        asm volatile("s_wait_asynccnt 0" ::: "memory");
#endif
    }
#else
    {
        // Portability fallback (no async builtin): plain staged copy.
        for (int k = 0; k < TILE / THREADS; ++k) {
            int idx = tid + k * THREADS;
            s_rows[idx] = rows[tileBase + idx];
            s_cols[idx] = cols[tileBase + idx];
            s_vals[idx] = vals[tileBase + idx];
        }
    }
#endif
    __syncthreads();

    const int wave    = tid >> 5;
    const int lane    = tid & 31;
    const int hopOff  = hop * D;
    const int nextOff = (hop + 1) * D;
    const unsigned char* em = emask + (size_t)hop * NNZ;

#pragma unroll 1
    for (int sub = 0; sub < (TILE / THREADS); ++sub) {
        int  sIdx = wave * 256 + sub * 32 + lane;
        long e    = tileBase + sIdx;
        int   r = s_rows[sIdx];
        int   c = s_cols[sIdx];
        float v = s_vals[sIdx];
        // Mask is streamed exactly once per hop: non-temporal so it doesn't
        // evict the L2-resident embedding state the gathers/atomics hit.
        bool keep = __builtin_nontemporal_load(em + e) != 0;
        unsigned long long bal = __ballot(keep);   // wave32: low 32 bits valid
        while (bal) {
            int j = __ffsll(bal) - 1;
            bal &= bal - 1;
            int   rj = __shfl(r, j, 32);
            int   cj = __shfl(c, j, 32);
            float vj = __shfl(v, j, 32) * EDGE_SCALE;
            // Gather source row (coalesced: 32 lanes x float2 = 256B)
            const float* src = out + (size_t)cj * OUT_STRIDE + hopOff;
            float2 x = *(const float2*)(src + lane * 2);
            // Scatter-add into destination row slice (L2 atomic units)
            float* dst = out + (size_t)rj * OUT_STRIDE + nextOff + lane * 2;
            atomicAdd(dst,     x.x * vj);
            atomicAdd(dst + 1, x.y * vj);
        }
    }
}

// ---------------------------------------------------------------------------
// Tail: remaining NNZ - FULL_TILES*TILE (=1664) edges, direct from global.
// ---------------------------------------------------------------------------
__global__ __launch_bounds__(THREADS) void lgcn_spmm_tail(
    const int* __restrict__ rows, const int* __restrict__ cols,
    const float* __restrict__ vals, const unsigned char* __restrict__ emask,
    float* __restrict__ out, int hop) {
    const int wave = threadIdx.x >> 5;
    const int lane = threadIdx.x & 31;
    const int hopOff  = hop * D;
    const int nextOff = (hop + 1) * D;
    const unsigned char* em = emask + (size_t)hop * NNZ;

    for (long base = TAIL_START + (long)wave * 32; base < (long)NNZ;
         base += (THREADS / 32) * 32) {
        long e = base + lane;
        int r = 0, c = 0;
        float v = 0.0f;
        bool keep = false;
        if (e < (long)NNZ) {
            r = rows[e];
            c = cols[e];
            v = vals[e];
            keep = em[e] != 0;
        }
        unsigned long long bal = __ballot(keep);
        while (bal) {
            int j = __ffsll(bal) - 1;
            bal &= bal - 1;
            int   rj = __shfl(r, j, 32);
            int   cj = __shfl(c, j, 32);
            float vj = __shfl(v, j, 32) * EDGE_SCALE;
            const float* src = out + (size_t)cj * OUT_STRIDE + hopOff;
            float2 x = *(const float2*)(src + lane * 2);
            float* dst = out + (size_t)rj * OUT_STRIDE + nextOff + lane * 2;
            atomicAdd(dst,     x.x * vj);
            atomicAdd(dst + 1, x.y * vj);
        }
    }
}

// ---------------------------------------------------------------------------
// Hop-0 init: out[node, 0, :] = concat(user, item); zero slices for hops 1..3.
// ---------------------------------------------------------------------------
__global__ void lgcn_init(const float* __restrict__ user_e,
                          const float* __restrict__ item_e,
                          float* __restrict__ out) {
    int i = blockIdx.x * blockDim.x + threadIdx.x;   // over N_NODES * D
    if (i >= N_NODES * D) return;
    int node = i >> 6;
    int d    = i & 63;
    float v = (i < N_USERS * D) ? __builtin_nontemporal_load(user_e + i)
                                : __builtin_nontemporal_load(item_e + (i - N_USERS * D));
    size_t base = (size_t)node * OUT_STRIDE + d;
    out[base]         = v;
    out[base + D]     = 0.0f;
    out[base + 2 * D] = 0.0f;
    out[base + 3 * D] = 0.0f;
}

// ---------------------------------------------------------------------------
// Message dropout: out[:, hop+1, :] = mask ? out * (1/0.9) : 0
// ---------------------------------------------------------------------------
__global__ void lgcn_mess(const unsigned char* __restrict__ mmask,
                          float* __restrict__ out, int hop) {
    long i = (long)blockIdx.x * blockDim.x + threadIdx.x;  // over N_NODES * D
    if (i >= (long)N_NODES * D) return;
    long node = i >> 6;
    int  d    = (int)(i & 63);
    size_t o  = (size_t)node * OUT_STRIDE + (size_t)(hop + 1) * D + d;
    bool keep = __builtin_nontemporal_load(mmask + (size_t)hop * N_NODES * D + i) != 0;
    float v   = out[o];
    out[o]    = keep ? v * MESS_SCALE : 0.0f;
}

// ---------------------------------------------------------------------------
extern "C" void kernel_launch(void* const* d_in, const int* in_sizes, int n_in,
                              void* d_out, int out_size, void* d_ws, size_t ws_size,
                              hipStream_t stream) {
    (void)in_sizes; (void)n_in; (void)out_size; (void)d_ws; (void)ws_size;
    const float*         user_e = (const float*)d_in[0];
    const float*         item_e = (const float*)d_in[1];
    const int*           rows   = (const int*)d_in[2];
    const int*           cols   = (const int*)d_in[3];
    const float*         vals   = (const float*)d_in[4];
    const unsigned char* emask  = (const unsigned char*)d_in[5];  // bool [3, NNZ]
    const unsigned char* mmask  = (const unsigned char*)d_in[6];  // bool [3, N, 64]
    float* out = (float*)d_out;                                   // [N, 4, 64]

    const int elemBlocks = (N_NODES * D + THREADS - 1) / THREADS; // 37500

    lgcn_init<<<elemBlocks, THREADS, 0, stream>>>(user_e, item_e, out);
    for (int hop = 0; hop < N_HOPS; ++hop) {
        lgcn_spmm<<<FULL_TILES, THREADS, 0, stream>>>(rows, cols, vals, emask, out, hop);
        lgcn_spmm_tail<<<1, THREADS, 0, stream>>>(rows, cols, vals, emask, out, hop);
        lgcn_mess<<<elemBlocks, THREADS, 0, stream>>>(mmask, out, hop);
    }
}